// Group_68470368632963
// MI455X (gfx1250) — compile-verified
//
#include <hip/hip_runtime.h>
#include <hip/hip_bf16.h>

// ---------------------------------------------------------------------------
// Problem: B=16, N=8192, C=3, G=512 groups, K=32 neighbors.
//   out = [neighborhood (B*G*K*3 fp32) | center (B*G*3 fp32)]
// Input fits in L2 (1.5MB) -> latency bound, fp32 kept to match reference
// top-k index selection. WMMA F32_16X16X4_F32 computes the c.x dot tiles.
// Selection uses a 32-wide ballot pre-filter (exact vs serial scan).
// ---------------------------------------------------------------------------

typedef __attribute__((ext_vector_type(2))) float v2f;
typedef __attribute__((ext_vector_type(8))) float v8f;

#define NPTS   8192
#define NGRP   512
#define KNN    32
#define NBATCH 16

// ======================= Kernel 1: farthest point sampling =================
// One block per batch. 1024 threads, 8 points/thread, all state in VGPRs.
__global__ __launch_bounds__(1024) void fps_kernel(const float* __restrict__ xyz,
                                                   float* __restrict__ center_out)
{
    const int b    = blockIdx.x;
    const int t    = threadIdx.x;
    const int lane = t & 31;
    const int wave = t >> 5;
    const float* base = xyz + (size_t)b * NPTS * 3;

    // strided ownership: point p = t + i*1024
    float px[8], py[8], pz[8], md[8];
#pragma unroll
    for (int i = 0; i < 8; ++i) {
        const int p = t + i * 1024;
        px[i] = base[p * 3 + 0];
        py[i] = base[p * 3 + 1];
        pz[i] = base[p * 3 + 2];
        md[i] = 1e10f;
    }

    __shared__ float s_cx, s_cy, s_cz;
    __shared__ float s_wd[32];
    __shared__ int   s_wi[32];
    __shared__ int   s_win;

    if (t == 0) {
        // fps convention: first center is point 0
        s_cx = base[0]; s_cy = base[1]; s_cz = base[2];
        float* co = center_out + (size_t)b * NGRP * 3;
        co[0] = s_cx; co[1] = s_cy; co[2] = s_cz;
    }
    __syncthreads();

    for (int g = 1; g < NGRP; ++g) {
        const float cx = s_cx, cy = s_cy, cz = s_cz;
        float bm = -1.0f;
        int   bi = 0x7fffffff;
#pragma unroll
        for (int i = 0; i < 8; ++i) {
            const float dx = px[i] - cx, dy = py[i] - cy, dz = pz[i] - cz;
            const float d = dx * dx + dy * dy + dz * dz;
            md[i] = fminf(md[i], d);
            const int pidx = t + i * 1024;
            if (md[i] > bm || (md[i] == bm && pidx < bi)) { bm = md[i]; bi = pidx; }
        }
        // wave32 argmax reduction (tie -> smaller index, matching jnp.argmax)
#pragma unroll
        for (int s = 16; s > 0; s >>= 1) {
            const float od = __shfl_xor(bm, s, 32);
            const int   oi = __shfl_xor(bi, s, 32);
            if (od > bm || (od == bm && oi < bi)) { bm = od; bi = oi; }
        }
        if (lane == 0) { s_wd[wave] = bm; s_wi[wave] = bi; }
        __syncthreads();
        if (t < 32) {
            float vd = s_wd[t];
            int   vi = s_wi[t];
#pragma unroll
            for (int s = 16; s > 0; s >>= 1) {
                const float od = __shfl_xor(vd, s, 32);
                const int   oi = __shfl_xor(vi, s, 32);
                if (od > vd || (od == vd && oi < vi)) { vd = od; vi = oi; }
            }
            if (t == 0) s_win = vi;
        }
        __syncthreads();
        if (t == 0) {
            const int win = s_win;
            const float* wp = base + (size_t)win * 3;  // L2-hot reload; avoids
            const float wx = wp[0], wy = wp[1], wz = wp[2];  // dynamic VGPR idx
            s_cx = wx; s_cy = wy; s_cz = wz;
            float* co = center_out + ((size_t)b * NGRP + g) * 3;
            co[0] = wx; co[1] = wy; co[2] = wz;
        }
        __syncthreads();
    }
}

// ======================= Kernel 2: kNN + gather via WMMA ===================
// 512 blocks (B x G/16), 512 threads = 16 waves. Block owns 16 centers.
// Per 256-point chunk: wave w computes a 16x16 dot tile with
// v_wmma_f32_16x16x4_f32, converts to ||c||^2+||x||^2-2c.x in LDS, then
// streams its center's row into an exact running top-32 (1 cand / lane)
// using a 32-wide ballot pre-filter.
__global__ __launch_bounds__(512) void knn_kernel(const float* __restrict__ xyz,
                                                  const float* __restrict__ centers,
                                                  float* __restrict__ out)
{
    const int b    = blockIdx.x >> 5;   // / (G/16)
    const int ct   = blockIdx.x & 31;
    const int c0   = ct * 16;
    const int t    = threadIdx.x;
    const int lane = t & 31;
    const int wave = t >> 5;

    const float* pbase = xyz + (size_t)b * NPTS * 3;
    const float* cbase = centers + ((size_t)b * NGRP + c0) * 3;

    __shared__ float s_cx[16], s_cy[16], s_cz[16], s_cc[16];
    __shared__ float s_dist[16][256];   // 16 KB distance tile

    if (t < 16) {
        const float x = cbase[t * 3 + 0];
        const float y = cbase[t * 3 + 1];
        const float z = cbase[t * 3 + 2];
        s_cx[t] = x; s_cy[t] = y; s_cz[t] = z;
        s_cc[t] = x * x + y * y + z * z;
    }
    __syncthreads();

    const int  m16 = lane & 15;
    const bool hi  = lane >= 16;

    // A fragment: 16x4 f32 (centers, K=3 zero-padded).  ISA layout:
    // lanes 0-15 hold (K=0,K=1), lanes 16-31 hold (K=2,K=3) for row M=lane&15.
    v2f afrag;
    afrag[0] = hi ? s_cz[m16] : s_cx[m16];
    afrag[1] = hi ? 0.0f      : s_cy[m16];

    // running top-32 state: one kept candidate per lane
    float bestD  = 3.0e38f;
    int   bestI  = 0x7fffffff;
    float curMax = 3.0e38f;
    int   maxLane = 31;

    for (int chunk = 0; chunk < NPTS / 256; ++chunk) {
        const int gidx = chunk * 256 + wave * 16 + m16;
        const float* pp = pbase + (size_t)gidx * 3;
        const float ptx = pp[0], pty = pp[1], ptz = pp[2];
        const float pn  = ptx * ptx + pty * pty + ptz * ptz;

        // B fragment: 4x16 f32 (points).  lanes 0-15: (K=0,K=1) rows,
        // lanes 16-31: (K=2,K=3) rows, column N=lane&15.
        v2f bfrag;
        bfrag[0] = hi ? ptz  : ptx;
        bfrag[1] = hi ? 0.0f : pty;

        v8f acc = {};
        acc = __builtin_amdgcn_wmma_f32_16x16x4_f32(
            /*neg_a=*/false, afrag, /*neg_b=*/false, bfrag,
            /*c_mod=*/(short)0, acc, /*reuse_a=*/false, /*reuse_b=*/false);

        // C/D layout: VGPR r, lane l -> (M = r + 8*(l>=16), N = l&15)
        const int col = wave * 16 + m16;
#pragma unroll
        for (int r = 0; r < 8; ++r) {
            const int M = r + (hi ? 8 : 0);
            s_dist[M][col] = s_cc[M] + pn - 2.0f * acc[r];
        }
        __syncthreads();

        // wave w scans row w, 32 candidates at a time. Ballot pre-filter is
        // exact: curMax only decreases, so anything failing the batch-start
        // test would also fail serially; survivors are re-checked in
        // ascending index order against the live curMax.
        for (int p0 = 0; p0 < 256; p0 += 32) {
            const float dl = s_dist[wave][p0 + lane];
            unsigned int mask = (unsigned int)__ballot(dl < curMax);
            while (mask) {
                const int j = __builtin_ctz(mask);  // ascending index order
                mask &= mask - 1;
                const float dj = __shfl(dl, j, 32); // wave-uniform broadcast
                if (dj < curMax) {
                    const int gi = chunk * 256 + p0 + j;
                    if (lane == maxLane) { bestD = dj; bestI = gi; }
                    // re-reduce wave max (tie: evict larger index first)
                    float rd = bestD; int ri = bestI; int rl = lane;
#pragma unroll
                    for (int s = 16; s > 0; s >>= 1) {
                        const float od = __shfl_xor(rd, s, 32);
                        const int   oi = __shfl_xor(ri, s, 32);
                        const int   ol = __shfl_xor(rl, s, 32);
                        if (od > rd ||
                            (od == rd && (oi > ri || (oi == ri && ol > rl)))) {
                            rd = od; ri = oi; rl = ol;
                        }
                    }
                    curMax = rd; maxLane = rl;
                }
            }
        }
        __syncthreads();
    }

    // in-wave bitonic sort of the 32 kept (dist, idx), ascending — matches
    // lax.top_k output ordering (ascending distance, index tiebreak)
    float d  = bestD;
    int   id = bestI;
#pragma unroll
    for (int k = 2; k <= 32; k <<= 1) {
#pragma unroll
        for (int j = k >> 1; j > 0; j >>= 1) {
            const float od = __shfl_xor(d, j, 32);
            const int   oi = __shfl_xor(id, j, 32);
            const bool up          = (lane & k) == 0;
            const bool iGreater    = (d > od) || (d == od && id > oi);
            const bool keepSmaller = ((lane & j) == 0) == up;
            const bool take = keepSmaller ? iGreater : !iGreater;
            if (take) { d = od; id = oi; }
        }
    }

    // gather + center-subtract: wave w -> center c0+w, lane k -> neighbor k
    const int   c  = c0 + wave;
    const float cx = s_cx[wave], cy = s_cy[wave], cz = s_cz[wave];
    const float* np = pbase + (size_t)id * 3;
    float* o = out + (((size_t)b * NGRP + c) * KNN + lane) * 3;
    o[0] = np[0] - cx;
    o[1] = np[1] - cy;
    o[2] = np[2] - cz;
}

// =========================== launch ========================================
extern "C" void kernel_launch(void* const* d_in, const int* in_sizes, int n_in,
                              void* d_out, int out_size, void* d_ws, size_t ws_size,
                              hipStream_t stream)
{
    (void)in_sizes; (void)n_in; (void)out_size; (void)d_ws; (void)ws_size;
    const float* xyz = (const float*)d_in[0];
    float* out = (float*)d_out;
    float* centers = out + (size_t)NBATCH * NGRP * KNN * 3;  // second output region

    fps_kernel<<<NBATCH, 1024, 0, stream>>>(xyz, centers);
    knn_kernel<<<NBATCH * (NGRP / 16), 512, 0, stream>>>(xyz, centers, out);
}